// MaskPyramids_52261162058458
// MI455X (gfx1250) — compile-verified
//
#include <hip/hip_runtime.h>

typedef __attribute__((ext_vector_type(2))) float v2f;
typedef __attribute__((ext_vector_type(8))) float v8f;

#define N_INST          512
#define ROWS_PER_INST   27       // 13 + 7 + 4 + 2 + 1 tile-rows
#define INST_STRIDE     53294    // 40000 + 10000 + 2500 + 625 + 169
#define GTAB            28
#define GCTR            13

// One 16x16 output tile = rank-1 outer product via V_WMMA_F32_16X16X4_F32.
// A (row factors) is hoisted per tile-row; only B changes per tile.
__device__ __forceinline__ v8f outer_tile(v2f A, const float* __restrict__ gmask,
                                          int sj, bool lo16)
{
    const int gj   = min(max(sj, 0), GTAB - 1);
    const float bv = gmask[GCTR * GTAB + gj];          // 28 hot floats
    const float bval = (lo16 && sj >= 0 && sj < GTAB) ? bv : 0.0f;
    v2f B; B.x = bval; B.y = 0.0f;                     // K=0 row in lanes 0-15
    v8f Cz = {0.0f, 0.0f, 0.0f, 0.0f, 0.0f, 0.0f, 0.0f, 0.0f};
    return __builtin_amdgcn_wmma_f32_16x16x4_f32(
        false, A, false, B, (short)0, Cz, false, false);
}

// One wave32 = one full tile-row (NTW tiles) of one (instance, level) mask.
// With constexpr H/NTW, every store in the row is an immediate-offset
// global_store_b32 off a single base address.
template <int H, int TRBASE, int EBASE>
__device__ __forceinline__ void level_row(
    int inst, int t, int lane,
    const int*   __restrict__ pos,
    const float* __restrict__ gmask,
    float*       __restrict__ out)
{
    constexpr int NTW = (H + 15) / 16;      // tiles per row/col
    const int tm = t - TRBASE;              // tile-row index

    // Level-space center (round-half-even matches jnp.round under RNE).
    const int p0 = pos[2 * inst + 0];
    const int p1 = pos[2 * inst + 1];
    const int lh = (int)__builtin_rintf((float)p0 / 200.0f * (float)H);
    const int lw = (int)__builtin_rintf((float)p1 / 200.0f * (float)H);

    const int  m16  = lane & 15;
    const bool lo16 = (lane < 16);

    // A factors (fixed for the whole tile-row): a[r] = valid ? gmask[si][13] : 0
    const int   si   = GCTR + tm * 16 + m16 - lh;
    const int   gi   = min(max(si, 0), GTAB - 1);
    const float av   = gmask[gi * GTAB + GCTR];
    const float aval = (lo16 && si >= 0 && si < GTAB) ? av : 0.0f;
    v2f A; A.x = aval; A.y = 0.0f;          // K=0 column in lanes 0-15

    const int sjb = GCTR + m16 - lw;        // per-tile: sj = sjb + tn*16

    // D layout: VGPR v, lanes 0-15 -> M=v, lanes 16-31 -> M=v+8; N = lane&15.
    const int mhi  = (lane >> 4) * 8;
    const int row0 = tm * 16 + mhi;
    // All flat offsets fit in int32 (27.3M elements total).
    float* p = out + inst * INST_STRIDE + EBASE + row0 * H + m16;

    const bool rowfull = ((tm + 1) * 16 <= H);          // wave-uniform
    if (__builtin_amdgcn_readfirstlane((int)rowfull)) {
        // Interior tile-row: all 16 rows valid.
#pragma unroll
        for (int tn = 0; tn < NTW - 1; ++tn) {
            v8f D = outer_tile(A, gmask, sjb + tn * 16, lo16);
#pragma unroll
            for (int v = 0; v < 8; ++v)
                p[tn * 16 + v * H] = D[v];              // immediate-offset store
        }
        {   // Last column tile: guard col only (uniform across the 8 stores).
            constexpr int tn = NTW - 1;
            v8f D = outer_tile(A, gmask, sjb + tn * 16, lo16);
            const bool colok = (tn * 16 + m16) < H;
#pragma unroll
            for (int v = 0; v < 8; ++v)
                if (colok) p[tn * 16 + v * H] = D[v];
        }
    } else {
        // Bottom tile-row: rows clipped at H (H%16 != 0 for every level).
#pragma unroll
        for (int tn = 0; tn < NTW; ++tn) {
            v8f D = outer_tile(A, gmask, sjb + tn * 16, lo16);
            const bool colok = (tn == NTW - 1) ? ((tn * 16 + m16) < H) : true;
#pragma unroll
            for (int v = 0; v < 8; ++v)
                if (colok & ((row0 + v) < H))
                    p[tn * 16 + v * H] = D[v];
        }
    }
}

__global__ __launch_bounds__(256) void mask_pyramid_wmma(
    const int*   __restrict__ pos,
    const float* __restrict__ gmask,
    float*       __restrict__ out)
{
    const int lane = threadIdx.x & 31;
    const int wid  = (int)((blockIdx.x * blockDim.x + threadIdx.x) >> 5);

    const int inst = wid / ROWS_PER_INST;               // constant-divisor div
    const int t    = wid - inst * ROWS_PER_INST;

    // Tile-row prefix per level: [0, 13, 20, 24, 26]. Level is wave-uniform.
    int lvl = (t >= 13) + (t >= 20) + (t >= 24) + (t >= 26);
    lvl = __builtin_amdgcn_readfirstlane(lvl);

    switch (lvl) {
    case 0:  level_row<200,  0,     0>(inst, t, lane, pos, gmask, out); break;
    case 1:  level_row<100, 13, 40000>(inst, t, lane, pos, gmask, out); break;
    case 2:  level_row< 50, 20, 50000>(inst, t, lane, pos, gmask, out); break;
    case 3:  level_row< 25, 24, 52500>(inst, t, lane, pos, gmask, out); break;
    default: level_row< 13, 26, 53125>(inst, t, lane, pos, gmask, out); break;
    }
}

extern "C" void kernel_launch(void* const* d_in, const int* in_sizes, int n_in,
                              void* d_out, int out_size, void* d_ws, size_t ws_size,
                              hipStream_t stream) {
    const int*   pos   = (const int*)d_in[0];     // (512, 2) int32
    const float* gmask = (const float*)d_in[1];   // (28, 28) f32
    float*       out   = (float*)d_out;           // (512, 53294) f32

    // 512 instances * 27 tile-rows = 13824 waves = exactly 1728 blocks of
    // 8 wave32s (no remainder -> EXEC all-1s at every WMMA).
    const int total_waves = N_INST * ROWS_PER_INST;
    const int threads     = 256;
    const int blocks      = (total_waves * 32) / threads;  // 1728

    mask_pyramid_wmma<<<blocks, threads, 0, stream>>>(pos, gmask, out);
}